// DTHGNN_1795296330249
// MI455X (gfx1250) — compile-verified
//
#include <hip/hip_runtime.h>

// Problem constants (match reference)
#define NB      100000   // nodes
#define EB      20000    // hyperedges
#define TSTEPS  12
#define CINW    16
#define HCH     64
#define NNZC    160000
#define PRED    12

typedef __attribute__((ext_vector_type(2))) float v2f;
typedef __attribute__((ext_vector_type(8))) float v8f;

static __device__ __forceinline__ v8f wmma4(v2f a, v2f b, v8f c) {
  // D = A(16x4 f32) * B(4x16 f32) + C(16x16 f32), full fp32 on the matrix pipe
  return __builtin_amdgcn_wmma_f32_16x16x4_f32(false, a, false, b, (short)0, c,
                                               false, false);
}

// ---------------- degree / normalization ----------------
__global__ __launch_bounds__(256) void k_count(const int* __restrict__ ni,
                                               const int* __restrict__ ei,
                                               float* __restrict__ deg,
                                               float* __restrict__ bc) {
  int j = blockIdx.x * 256 + threadIdx.x;
  if (j >= TSTEPS * NNZC) return;
  int t = j / NNZC;
  atomicAdd(&deg[t * NB + ni[j]], 1.0f);
  atomicAdd(&bc[t * EB + ei[j]], 1.0f);
}

__global__ __launch_bounds__(256) void k_rsqrt_ip(float* __restrict__ d, int n) {
  int i = blockIdx.x * 256 + threadIdx.x;
  if (i < n) { float v = d[i]; d[i] = v > 0.f ? rsqrtf(v) : 0.f; }
}

__global__ __launch_bounds__(256) void k_recip_ip(float* __restrict__ d, int n) {
  int i = blockIdx.x * 256 + threadIdx.x;
  if (i < n) { float v = d[i]; d[i] = v > 0.f ? (1.0f / v) : 0.f; }
}

// ---------------- generic WMMA GEMM: out[rows][64] = A[rows][K] @ W[K][64] + b ----
// B is staged in LDS pre-paired along K: Bl[k/2][col] = {B[k][col], B[k+1][col]}
// so each WMMA B-operand is one aligned ds_load_b64.
__global__ __launch_bounds__(256) void k_gemm_bias(const float* __restrict__ A,
                                                   const float* __restrict__ W,
                                                   const float* __restrict__ bias,
                                                   float* __restrict__ out,
                                                   int rows, int K) {
  __shared__ v2f Bl[32 * 64];  // up to K=64 -> 32 pairs x 64 cols
  int tid = threadIdx.x;
  int pairs = K >> 1;
  for (int i = tid; i < pairs * 64; i += 256) {
    int h = i & 63, kp = i >> 6;
    v2f w = {W[(2 * kp) * HCH + h], W[(2 * kp + 1) * HCH + h]};
    Bl[i] = w;
  }
  __syncthreads();
  int lane = tid & 31, wv = tid >> 5;
  int half = lane >> 4, l16 = lane & 15;
  int rowbase = (blockIdx.x * 8 + wv) * 16;
  if (rowbase >= rows) return;
  const float* Ar = A + (size_t)(rowbase + l16) * K;
  v8f zero = {0.f, 0.f, 0.f, 0.f, 0.f, 0.f, 0.f, 0.f};
  v8f acc[4];
#pragma unroll
  for (int i = 0; i < 4; ++i) acc[i] = zero;
  for (int kb = 0; kb < K; kb += 4) {
    int k0 = kb + 2 * half;                 // even
    v2f a = *(const v2f*)(Ar + k0);
    const v2f* brow = &Bl[(k0 >> 1) * 64 + l16];
#pragma unroll
    for (int nt = 0; nt < 4; ++nt) {
      acc[nt] = wmma4(a, brow[nt * 16], acc[nt]);
    }
  }
#pragma unroll
  for (int nt = 0; nt < 4; ++nt) {
    int col = nt * 16 + l16;
    float bv = bias[col];
#pragma unroll
    for (int v = 0; v < 8; ++v)
      out[(size_t)(rowbase + v + 8 * half) * HCH + col] = acc[nt][v] + bv;
  }
}

// ---------------- hypergraph scatter kernels ----------------
__global__ __launch_bounds__(256) void k_scatter_fwd(const int* __restrict__ ni,
                                                     const int* __restrict__ ei,
                                                     const float* __restrict__ dv,
                                                     const float* __restrict__ src,
                                                     float* __restrict__ acc) {
  int g = blockIdx.x * 256 + threadIdx.x;
  if (g >= NNZC * 16) return;
  int j = g >> 4, c = (g & 15) << 2;
  int n = ni[j], e = ei[j];
  float s = dv[n];
  float4 x = *(const float4*)(src + (size_t)n * HCH + c);
  float* d = acc + (size_t)e * HCH + c;
  atomicAdd(d + 0, x.x * s);
  atomicAdd(d + 1, x.y * s);
  atomicAdd(d + 2, x.z * s);
  atomicAdd(d + 3, x.w * s);
}

__global__ __launch_bounds__(256) void k_scatter_back(const int* __restrict__ ni,
                                                      const int* __restrict__ ei,
                                                      const float* __restrict__ edge,
                                                      float* __restrict__ acc) {
  int g = blockIdx.x * 256 + threadIdx.x;
  if (g >= NNZC * 16) return;
  int j = g >> 4, c = (g & 15) << 2;
  int n = ni[j], e = ei[j];
  float4 x = *(const float4*)(edge + (size_t)e * HCH + c);
  float* d = acc + (size_t)n * HCH + c;
  atomicAdd(d + 0, x.x);
  atomicAdd(d + 1, x.y);
  atomicAdd(d + 2, x.z);
  atomicAdd(d + 3, x.w);
}

__global__ __launch_bounds__(256) void k_scale_rows(const float* __restrict__ src,
                                                    const float* __restrict__ sc,
                                                    float* __restrict__ dst, int rows) {
  int g = blockIdx.x * 256 + threadIdx.x;
  if (g >= rows * 16) return;
  int r = g >> 4, c = (g & 15) << 2;
  float s = sc[r];
  float4 x = *(const float4*)(src + (size_t)r * HCH + c);
  *(float4*)(dst + (size_t)r * HCH + c) =
      make_float4(x.x * s, x.y * s, x.z * s, x.w * s);
}

// ---------------- aggregator conv (WMMA, K=192) ----------------
// out[h][row] = bias[h] + sum_{hin,k} src[tt+k][n][hin] * Wc[h][hin][k]
// rows = n*9+tt, stored flat [64][R*9]  (== torch-faithful reshape [R][576])
__global__ __launch_bounds__(256) void k_aggconv(const float* __restrict__ src, int R,
                                                 const float* __restrict__ Wc,
                                                 const float* __restrict__ bias,
                                                 float* __restrict__ outf) {
  __shared__ v2f Bl[96 * 64];  // 192 K pre-paired
  int tid = threadIdx.x;
  for (int i = tid; i < 96 * 64; i += 256) {
    int h = i & 63, kp = i >> 6;
    int kidx = kp << 1;               // even; pair shares kk
    int hin = kidx & 63, kk = kidx >> 6;
    v2f w = {Wc[h * 192 + hin * 3 + kk], Wc[h * 192 + (hin + 1) * 3 + kk]};
    Bl[i] = w;
  }
  __syncthreads();
  int lane = tid & 31, wv = tid >> 5;
  int half = lane >> 4, l16 = lane & 15;
  int totalRows = R * 9;
  int tilebase = (blockIdx.x * 8 + wv) * 16;
  if (tilebase >= totalRows) return;
  int row = tilebase + l16;
  int n = row / 9;
  int tt = row - n * 9;
  v8f zero = {0.f, 0.f, 0.f, 0.f, 0.f, 0.f, 0.f, 0.f};
  v8f acc[4];
#pragma unroll
  for (int i = 0; i < 4; ++i) acc[i] = zero;
  for (int kb = 0; kb < 192; kb += 4) {
    int k0 = kb + 2 * half;           // even
    int t = tt + (k0 >> 6);
    int hin = k0 & 63;
    v2f a = *(const v2f*)(src + ((size_t)t * R + n) * HCH + hin);
    const v2f* brow = &Bl[(k0 >> 1) * 64 + l16];
#pragma unroll
    for (int nt = 0; nt < 4; ++nt) {
      acc[nt] = wmma4(a, brow[nt * 16], acc[nt]);
    }
  }
  size_t R9 = (size_t)totalRows;
#pragma unroll
  for (int nt = 0; nt < 4; ++nt) {
    int col = nt * 16 + l16;
    float bv = bias[col];
    float* p = outf + (size_t)col * R9 + tilebase + 8 * half;
    *(float4*)(p + 0) = make_float4(acc[nt][0] + bv, acc[nt][1] + bv,
                                    acc[nt][2] + bv, acc[nt][3] + bv);
    *(float4*)(p + 4) = make_float4(acc[nt][4] + bv, acc[nt][5] + bv,
                                    acc[nt][6] + bv, acc[nt][7] + bv);
  }
}

// ---------------- score MLP (WMMA, K=576, fused gather+product+2nd layer) -------
__global__ __launch_bounds__(256) void k_mlp(const float* __restrict__ aggf,
                                             const float* __restrict__ eaggf,
                                             const int* __restrict__ ni_last,
                                             const int* __restrict__ ei_last,
                                             const int* __restrict__ nni,
                                             const int* __restrict__ nei,
                                             const float* __restrict__ Wm1,
                                             const float* __restrict__ bm1,
                                             const float* __restrict__ Wm2,
                                             const float* __restrict__ bm2,
                                             float* __restrict__ out) {
  __shared__ v2f Bl[96 * 64];  // one 192-K chunk of Wm1, pre-paired
  __shared__ float w2s[64];
  __shared__ float b1s[64];
  int tid = threadIdx.x;
  if (tid < 64) { w2s[tid] = Wm2[tid]; b1s[tid] = bm1[tid]; }
  int lane = tid & 31, wv = tid >> 5;
  int half = lane >> 4, l16 = lane & 15;
  int tilebase = (blockIdx.x * 8 + wv) * 16;   // 2*NNZ rows, exact tiling
  int g = tilebase + l16;
  int n, e;
  if (g < NNZC) { n = ni_last[g]; e = ei_last[g]; }
  else          { n = nni[g - NNZC]; e = nei[g - NNZC]; }
  const float* ap = aggf  + (size_t)n * 576;
  const float* ep = eaggf + (size_t)e * 576;
  v8f zero = {0.f, 0.f, 0.f, 0.f, 0.f, 0.f, 0.f, 0.f};
  v8f acc[4];
#pragma unroll
  for (int i = 0; i < 4; ++i) acc[i] = zero;
  for (int ch = 0; ch < 3; ++ch) {
    __syncthreads();
    for (int i = tid; i < 96 * 64; i += 256) {
      int h = i & 63, kp = i >> 6;
      int krow = ch * 192 + (kp << 1);
      v2f w = {Wm1[krow * HCH + h], Wm1[(krow + 1) * HCH + h]};
      Bl[i] = w;
    }
    __syncthreads();
    for (int kb = 0; kb < 192; kb += 4) {
      int k0 = kb + 2 * half;          // even
      int kg = ch * 192 + k0;
      v2f x = *(const v2f*)(ap + kg);
      v2f y = *(const v2f*)(ep + kg);
      v2f a = x * y;
      const v2f* brow = &Bl[(k0 >> 1) * 64 + l16];
#pragma unroll
      for (int nt = 0; nt < 4; ++nt) {
        acc[nt] = wmma4(a, brow[nt * 16], acc[nt]);
      }
    }
  }
  // layer 2: out[row] = sum_h relu(acc + bm1[h]) * Wm2[h] + bm2
  float part[8];
#pragma unroll
  for (int v = 0; v < 8; ++v) {
    float p = 0.f;
#pragma unroll
    for (int nt = 0; nt < 4; ++nt) {
      int h = nt * 16 + l16;
      float hv = acc[nt][v] + b1s[h];
      hv = hv > 0.f ? hv : 0.f;
      p += hv * w2s[h];
    }
#pragma unroll
    for (int m = 1; m < 16; m <<= 1) p += __shfl_xor(p, m, 32);
    part[v] = p;
  }
  if (l16 == 0) {
    float bb = bm2[0];
    int rowb = tilebase + 8 * half;
#pragma unroll
    for (int v = 0; v < 8; ++v) out[NB * PRED + rowb + v] = part[v] + bb;
  }
}

// ---------------- temporal conv + residual + final conv (only h=61..63 live) ----
__global__ __launch_bounds__(256) void k_timeconv(const float* __restrict__ xs,
                                                  const float* __restrict__ nf,
                                                  const float* __restrict__ Wt,
                                                  const float* __restrict__ bt,
                                                  const float* __restrict__ Wr,
                                                  const float* __restrict__ br,
                                                  const float* __restrict__ Wf,
                                                  const float* __restrict__ bf,
                                                  float* __restrict__ out) {
  __shared__ float Wts[3 * 192];  // [i][hin*3+k], i -> h_out = 61+i
  __shared__ float Wrs[3 * 16];
  __shared__ float Wfs[PRED * TSTEPS * 3];
  int tid = threadIdx.x;
  for (int i = tid; i < 3 * 192; i += 256)
    Wts[i] = Wt[(61 + i / 192) * 192 + (i % 192)];
  for (int i = tid; i < 48; i += 256)
    Wrs[i] = Wr[(61 + i / 16) * 16 + (i % 16)];
  for (int i = tid; i < PRED * TSTEPS * 3; i += 256) Wfs[i] = Wf[i];
  __syncthreads();
  int n = blockIdx.x * 256 + tid;
  if (n >= NB) return;
  float xp[3][TSTEPS];
#pragma unroll
  for (int i = 0; i < 3; ++i)
    for (int t = 0; t < TSTEPS; ++t) xp[i][t] = 0.f;
  for (int tp = 0; tp < TSTEPS; ++tp) {
    const float* r = xs + ((size_t)tp * NB + n) * HCH;
    float s[3][3] = {{0.f}};
    for (int c = 0; c < HCH; c += 4) {
      float4 x = *(const float4*)(r + c);
#pragma unroll
      for (int i = 0; i < 3; ++i) {
        const float* w = &Wts[i * 192 + c * 3];
#pragma unroll
        for (int k = 0; k < 3; ++k)
          s[i][k] += x.x * w[k] + x.y * w[3 + k] + x.z * w[6 + k] + x.w * w[9 + k];
      }
    }
#pragma unroll
    for (int i = 0; i < 3; ++i)
#pragma unroll
      for (int k = 0; k < 3; ++k) {
        int t = tp + 1 - k;
        if (t >= 0 && t < TSTEPS) xp[i][t] += s[i][k];
      }
  }
  float o[PRED];
#pragma unroll
  for (int p = 0; p < PRED; ++p) o[p] = bf[p];
  for (int t = 0; t < TSTEPS; ++t) {
    const float* q = nf + ((size_t)t * NB + n) * CINW;
    float4 q0 = *(const float4*)(q);
    float4 q1 = *(const float4*)(q + 4);
    float4 q2 = *(const float4*)(q + 8);
    float4 q3 = *(const float4*)(q + 12);
#pragma unroll
    for (int i = 0; i < 3; ++i) {
      const float* w = &Wrs[i * 16];
      float rs = br[61 + i] + q0.x * w[0] + q0.y * w[1] + q0.z * w[2] + q0.w * w[3] +
                 q1.x * w[4] + q1.y * w[5] + q1.z * w[6] + q1.w * w[7] +
                 q2.x * w[8] + q2.y * w[9] + q2.z * w[10] + q2.w * w[11] +
                 q3.x * w[12] + q3.y * w[13] + q3.z * w[14] + q3.w * w[15];
      float v = xp[i][t] + bt[61 + i] + rs;
      v = v > 0.f ? v : 0.f;
#pragma unroll
      for (int p = 0; p < PRED; ++p) o[p] += v * Wfs[p * 36 + t * 3 + i];
    }
  }
#pragma unroll
  for (int p = 0; p < PRED; ++p) out[(size_t)n * PRED + p] = o[p];
}

// ---------------- host launcher ----------------
extern "C" void kernel_launch(void* const* d_in, const int* in_sizes, int n_in,
                              void* d_out, int out_size, void* d_ws, size_t ws_size,
                              hipStream_t stream) {
  const float* nf  = (const float*)d_in[0];
  const float* W1  = (const float*)d_in[1];
  const float* b1  = (const float*)d_in[2];
  const float* W2  = (const float*)d_in[3];
  const float* b2  = (const float*)d_in[4];
  const float* Wt  = (const float*)d_in[5];
  const float* bt  = (const float*)d_in[6];
  const float* Wr  = (const float*)d_in[7];
  const float* br  = (const float*)d_in[8];
  const float* Wf  = (const float*)d_in[9];
  const float* bf  = (const float*)d_in[10];
  const float* Wna = (const float*)d_in[11];
  const float* bna = (const float*)d_in[12];
  const float* Wea = (const float*)d_in[13];
  const float* bea = (const float*)d_in[14];
  const float* Wm1 = (const float*)d_in[15];
  const float* bm1 = (const float*)d_in[16];
  const float* Wm2 = (const float*)d_in[17];
  const float* bm2 = (const float*)d_in[18];
  const int* node_idx = (const int*)d_in[19];
  const int* edge_idx = (const int*)d_in[20];
  const int* nni = (const int*)d_in[21];
  const int* nei = (const int*)d_in[22];
  float* outp = (float*)d_out;

  // workspace partition (~738 MB total)
  float* ws = (float*)d_ws;
  float* dinv = ws;  ws += (size_t)TSTEPS * NB;
  float* binv = ws;  ws += (size_t)TSTEPS * EB;
  float* xw   = ws;  ws += (size_t)NB * HCH;
  float* h1   = ws;  ws += (size_t)NB * HCH;
  float* eacc = ws;  ws += (size_t)EB * HCH;
  float* efin = ws;  ws += (size_t)EB * HCH;
  float* nacc = ws;  ws += (size_t)NB * HCH;
  float* xs   = ws;  ws += (size_t)TSTEPS * NB * HCH;
  float* es   = ws;  ws += (size_t)TSTEPS * EB * HCH;
  float* aggf = ws;  ws += (size_t)HCH * NB * 9;
  float* eaggf= ws;  ws += (size_t)HCH * EB * 9;

  // degree normalizations
  hipMemsetAsync(dinv, 0, sizeof(float) * TSTEPS * NB, stream);
  hipMemsetAsync(binv, 0, sizeof(float) * TSTEPS * EB, stream);
  k_count<<<(TSTEPS * NNZC + 255) / 256, 256, 0, stream>>>(node_idx, edge_idx, dinv, binv);
  k_rsqrt_ip<<<(TSTEPS * NB + 255) / 256, 256, 0, stream>>>(dinv, TSTEPS * NB);
  k_recip_ip<<<(TSTEPS * EB + 255) / 256, 256, 0, stream>>>(binv, TSTEPS * EB);

  const int gemmBlocksN = (NB / 16 + 7) / 8;
  const int scatBlocks  = (NNZC * 16) / 256;
  const int scaleE      = (EB * 16) / 256;
  const int scaleN      = (NB * 16) / 256;

  for (int t = 0; t < TSTEPS; ++t) {
    const int* ni = node_idx + (size_t)t * NNZC;
    const int* ei = edge_idx + (size_t)t * NNZC;
    const float* dv = dinv + (size_t)t * NB;
    const float* bv = binv + (size_t)t * EB;
    float* est = es + (size_t)t * EB * HCH;
    float* xst = xs + (size_t)t * NB * HCH;

    // hconv round 1: xw = x_t @ W1 + b1
    k_gemm_bias<<<gemmBlocksN, 256, 0, stream>>>(nf + (size_t)t * NB * CINW, W1, b1,
                                                 xw, NB, CINW);
    hipMemsetAsync(eacc, 0, sizeof(float) * EB * HCH, stream);
    k_scatter_fwd<<<scatBlocks, 256, 0, stream>>>(ni, ei, dv, xw, eacc);
    k_scale_rows<<<scaleE, 256, 0, stream>>>(eacc, bv, efin, EB);
    hipMemsetAsync(nacc, 0, sizeof(float) * NB * HCH, stream);
    k_scatter_back<<<scatBlocks, 256, 0, stream>>>(ni, ei, efin, nacc);
    k_scale_rows<<<scaleN, 256, 0, stream>>>(nacc, dv, h1, NB);

    // hconv round 2: xw = h1 @ W2 + b2 ; keep edge emb (es) and node out (xs)
    k_gemm_bias<<<gemmBlocksN, 256, 0, stream>>>(h1, W2, b2, xw, NB, HCH);
    hipMemsetAsync(eacc, 0, sizeof(float) * EB * HCH, stream);
    k_scatter_fwd<<<scatBlocks, 256, 0, stream>>>(ni, ei, dv, xw, eacc);
    k_scale_rows<<<scaleE, 256, 0, stream>>>(eacc, bv, est, EB);
    hipMemsetAsync(nacc, 0, sizeof(float) * NB * HCH, stream);
    k_scatter_back<<<scatBlocks, 256, 0, stream>>>(ni, ei, est, nacc);
    k_scale_rows<<<scaleN, 256, 0, stream>>>(nacc, dv, xst, NB);
  }

  // aggregator convs (WMMA)
  k_aggconv<<<(NB * 9 / 16 + 7) / 8, 256, 0, stream>>>(xs, NB, Wna, bna, aggf);
  k_aggconv<<<(EB * 9 / 16 + 7) / 8, 256, 0, stream>>>(es, EB, Wea, bea, eaggf);

  // temporal conv + residual + final conv -> indiv_logit [N,12]
  k_timeconv<<<(NB + 255) / 256, 256, 0, stream>>>(xs, nf, Wt, bt, Wr, br, Wf, bf, outp);

  // score MLP -> pos/neg logits (2*NNZ rows, exact 16-row tiling)
  k_mlp<<<(2 * NNZC / 16) / 8, 256, 0, stream>>>(aggf, eaggf,
                                                 node_idx + (size_t)(TSTEPS - 1) * NNZC,
                                                 edge_idx + (size_t)(TSTEPS - 1) * NNZC,
                                                 nni, nei, Wm1, bm1, Wm2, bm2, outp);
}